// GraphEncoder_32916629356847
// MI455X (gfx1250) — compile-verified
//
#include <hip/hip_runtime.h>

// ---------------------------------------------------------------------------
// GCN 3-layer encoder for MI455X (gfx1250, wave32).
// GEMMs: V_WMMA_F32_16X16X4_F32, A-tile staged in LDS (8x reuse across waves),
// K fully unrolled via template. Scatter: float4 gather + global_atomic_add_f32.
// Fused BN+ReLU; fused LayerNorm + mean-pool with wave32 shfl reductions.
// ---------------------------------------------------------------------------

#define N_NODES 100000
#define N_EDGES 1600000
#define G_GRAPHS 64
#define IN_C  64
#define HID_C 128
#define OUT_C 64
#define EPS_F 1e-5f

typedef float v2f __attribute__((ext_vector_type(2)));
typedef float v8f __attribute__((ext_vector_type(8)));

// ---------------- degree / init ----------------
__global__ void k_init(float* __restrict__ deg, float* __restrict__ psum,
                       float* __restrict__ pcnt) {
  int i = blockIdx.x * blockDim.x + threadIdx.x;
  if (i < N_NODES) deg[i] = 1.0f;               // self-loop
  if (i < G_GRAPHS * OUT_C) psum[i] = 0.0f;
  if (i < G_GRAPHS) pcnt[i] = 0.0f;
}

__global__ void k_count_deg(float* __restrict__ deg, const int* __restrict__ dst) {
  int e = blockIdx.x * blockDim.x + threadIdx.x;
  if (e < N_EDGES) atomicAdd(&deg[dst[e]], 1.0f);
}

__global__ void k_dinv(float* __restrict__ dinv, const float* __restrict__ deg) {
  int i = blockIdx.x * blockDim.x + threadIdx.x;
  if (i < N_NODES) dinv[i] = rsqrtf(deg[i]);
}

// ---------------- GEMM: Out[M,COUT] = A[M,K] @ W[K,COUT] via WMMA f32 -------
// One block = one 16-row M-tile; wave w computes N-tile w. The 16xK A-tile is
// contiguous in memory (tile spans full K), so stage it in LDS once per block
// and let all COUT/16 waves read fragments from LDS (8x traffic reduction).
// K is a template constant -> full unroll, back-to-back v_wmma chains.
// No divergence anywhere: EXEC all-1s as WMMA requires.
template <int K, int COUT>
__global__ void k_gemm_wmma(const float* __restrict__ A, const float* __restrict__ W,
                            float* __restrict__ Out) {
  __shared__ float As[16 * K];
  const int tid  = threadIdx.x;
  const int nthr = 32 * (COUT / 16);
  const int m0   = blockIdx.x * 16;

  // cooperative contiguous float4 copy of the 16xK tile into LDS
  const float4* Ag  = (const float4*)(A + (size_t)m0 * K);
  float4*       As4 = (float4*)As;
#pragma unroll
  for (int i = tid; i < (16 * K) / 4; i += nthr) As4[i] = Ag[i];
  __syncthreads();

  const int lane = tid & 31;
  const int wave = tid >> 5;             // N-tile index
  const int r    = lane & 15;            // A-row / B-col / D-col (lane % 16)
  const int kh   = lane >> 4;            // K-half select (0 or 1)
  const int n0   = wave * 16;

  v8f acc = {0.f, 0.f, 0.f, 0.f, 0.f, 0.f, 0.f, 0.f};
#pragma unroll
  for (int kk = 0; kk < K; kk += 4) {
    const int ka = kk + kh * 2;
    v2f a, b;
    a.x = As[r * K + ka];                 // ds_load from staged tile
    a.y = As[r * K + ka + 1];
    b.x = W[ka * COUT + n0 + r];          // W is tiny -> L2/WGP$ resident
    b.y = W[(ka + 1) * COUT + n0 + r];
    acc = __builtin_amdgcn_wmma_f32_16x16x4_f32(
        /*neg_a=*/false, a, /*neg_b=*/false, b,
        /*c_mod=*/(short)0, acc, /*reuse_a=*/false, /*reuse_b=*/false);
  }

  float* o = Out + (size_t)(m0 + kh * 8) * COUT + n0 + r;
#pragma unroll
  for (int v = 0; v < 8; ++v) o[(size_t)v * COUT] = acc[v];
}

// ---------------- self-loop term: agg = xw / deg ----------------------------
__global__ void k_self(float* __restrict__ agg, const float* __restrict__ xw,
                       const float* __restrict__ deg, int C) {
  int i = blockIdx.x * blockDim.x + threadIdx.x;
  if (i >= N_NODES * C) return;
  int n = i / C;
  agg[i] = xw[i] / deg[n];
}

// ---------------- edge scatter: agg[dst] += xw[src] * dinv[s]*dinv[d] -------
// One thread per (edge, 4 channels): float4 gather + 4 float atomics.
__global__ void k_scatter(float* __restrict__ agg, const float* __restrict__ xw,
                          const float* __restrict__ dinv,
                          const int* __restrict__ src, const int* __restrict__ dst,
                          int C4, int C) {
  int idx = blockIdx.x * blockDim.x + threadIdx.x;
  if (idx >= N_EDGES * C4) return;
  int e  = idx / C4;
  int c4 = idx - e * C4;
  int s = src[e], d = dst[e];
  float nrm = dinv[s] * dinv[d];
  float4 v = ((const float4*)(xw + (size_t)s * C))[c4];
  float* o = agg + (size_t)d * C + c4 * 4;
  atomicAdd(o + 0, v.x * nrm);
  atomicAdd(o + 1, v.y * nrm);
  atomicAdd(o + 2, v.z * nrm);
  atomicAdd(o + 3, v.w * nrm);
}

// ---------------- fused bias + BatchNorm(eval) + ReLU -----------------------
__global__ void k_bn_relu(float* __restrict__ out, const float* __restrict__ agg,
                          const float* __restrict__ bias,
                          const float* __restrict__ g, const float* __restrict__ bb,
                          const float* __restrict__ m, const float* __restrict__ v,
                          int C) {
  int i = blockIdx.x * blockDim.x + threadIdx.x;
  if (i >= N_NODES * C) return;
  int c = i % C;
  float h = agg[i] + bias[c];
  h = (h - m[c]) * rsqrtf(v[c] + EPS_F) * g[c] + bb[c];
  out[i] = fmaxf(h, 0.0f);
}

// ---------------- fused bias + LayerNorm + mean-pool accumulate -------------
// One wave per node; 64 channels = 2 per lane; wave32 shfl_xor reductions.
__global__ void k_ln_pool(float* __restrict__ node_out, float* __restrict__ psum,
                          float* __restrict__ pcnt,
                          const float* __restrict__ agg, const float* __restrict__ bias,
                          const float* __restrict__ lng, const float* __restrict__ lnb,
                          const int* __restrict__ batch) {
  const int lane = threadIdx.x & 31;
  const int wave = threadIdx.x >> 5;
  const int n = blockIdx.x * (blockDim.x >> 5) + wave;
  if (n >= N_NODES) return;

  const float* row = agg + (size_t)n * OUT_C;
  float h0 = row[lane] + bias[lane];
  float h1 = row[lane + 32] + bias[lane + 32];

  float s = h0 + h1;
#pragma unroll
  for (int off = 16; off > 0; off >>= 1) s += __shfl_xor(s, off, 32);
  float mu = s * (1.0f / OUT_C);

  float d0 = h0 - mu, d1 = h1 - mu;
  float vs = d0 * d0 + d1 * d1;
#pragma unroll
  for (int off = 16; off > 0; off >>= 1) vs += __shfl_xor(vs, off, 32);
  float rstd = rsqrtf(vs * (1.0f / OUT_C) + EPS_F);

  float y0 = d0 * rstd * lng[lane] + lnb[lane];
  float y1 = d1 * rstd * lng[lane + 32] + lnb[lane + 32];

  float* orow = node_out + (size_t)n * OUT_C;
  orow[lane] = y0;
  orow[lane + 32] = y1;

  int gidx = batch[n];
  atomicAdd(&psum[gidx * OUT_C + lane], y0);
  atomicAdd(&psum[gidx * OUT_C + lane + 32], y1);
  if (lane == 0) atomicAdd(&pcnt[gidx], 1.0f);
}

__global__ void k_pool_final(float* __restrict__ graph_out,
                             const float* __restrict__ psum,
                             const float* __restrict__ pcnt) {
  int i = blockIdx.x * blockDim.x + threadIdx.x;
  if (i >= G_GRAPHS * OUT_C) return;
  graph_out[i] = psum[i] / fmaxf(pcnt[i / OUT_C], 1.0f);
}

// ---------------------------------------------------------------------------
extern "C" void kernel_launch(void* const* d_in, const int* in_sizes, int n_in,
                              void* d_out, int out_size, void* d_ws, size_t ws_size,
                              hipStream_t stream) {
  const float* x     = (const float*)d_in[0];
  const int*   ei    = (const int*)d_in[1];     // [2,E]: row0=src, row1=dst
  const int*   batch = (const int*)d_in[2];
  const float* W0 = (const float*)d_in[3];
  const float* b0 = (const float*)d_in[4];
  const float* W1 = (const float*)d_in[5];
  const float* b1 = (const float*)d_in[6];
  const float* W2 = (const float*)d_in[7];
  const float* b2 = (const float*)d_in[8];
  const float* bn0_g = (const float*)d_in[9];
  const float* bn0_b = (const float*)d_in[10];
  const float* bn0_m = (const float*)d_in[11];
  const float* bn0_v = (const float*)d_in[12];
  const float* bn1_g = (const float*)d_in[13];
  const float* bn1_b = (const float*)d_in[14];
  const float* bn1_m = (const float*)d_in[15];
  const float* bn1_v = (const float*)d_in[16];
  const float* ln_g  = (const float*)d_in[17];
  const float* ln_b  = (const float*)d_in[18];

  const int* src = ei;
  const int* dst = ei + N_EDGES;

  float* node_out  = (float*)d_out;
  float* graph_out = node_out + (size_t)N_NODES * OUT_C;

  // workspace partition (floats): 2x big ping-pong + deg + dinv + pool
  float* ws   = (float*)d_ws;
  const size_t big = (size_t)N_NODES * HID_C;
  float* bufA = ws;
  float* bufB = bufA + big;
  float* deg  = bufB + big;
  float* dinv = deg + N_NODES;
  float* psum = dinv + N_NODES;
  float* pcnt = psum + (size_t)G_GRAPHS * OUT_C;

  const int B = 256;
  const int gN    = (N_NODES + B - 1) / B;
  const int gE    = (N_EDGES + B - 1) / B;
  const int gNC_h = (N_NODES * HID_C + B - 1) / B;
  const int gNC_o = (N_NODES * OUT_C + B - 1) / B;
  const int gS_h  = (N_EDGES * (HID_C / 4) + B - 1) / B;
  const int gS_o  = (N_EDGES * (OUT_C / 4) + B - 1) / B;
  const int mTiles = N_NODES / 16;           // 6250 exact

  // degree + pooling init
  k_init<<<gN, B, 0, stream>>>(deg, psum, pcnt);
  k_count_deg<<<gE, B, 0, stream>>>(deg, dst);
  k_dinv<<<gN, B, 0, stream>>>(dinv, deg);

  // ---- layer 0: x(64) -> 128 ----
  k_gemm_wmma<IN_C, HID_C><<<mTiles, 32 * (HID_C / 16), 0, stream>>>(x, W0, bufA);
  k_self<<<gNC_h, B, 0, stream>>>(bufB, bufA, deg, HID_C);
  k_scatter<<<gS_h, B, 0, stream>>>(bufB, bufA, dinv, src, dst, HID_C / 4, HID_C);
  k_bn_relu<<<gNC_h, B, 0, stream>>>(bufA, bufB, b0, bn0_g, bn0_b, bn0_m, bn0_v, HID_C);

  // ---- layer 1: 128 -> 128 ----
  k_gemm_wmma<HID_C, HID_C><<<mTiles, 32 * (HID_C / 16), 0, stream>>>(bufA, W1, bufB);
  k_self<<<gNC_h, B, 0, stream>>>(bufA, bufB, deg, HID_C);
  k_scatter<<<gS_h, B, 0, stream>>>(bufA, bufB, dinv, src, dst, HID_C / 4, HID_C);
  k_bn_relu<<<gNC_h, B, 0, stream>>>(bufB, bufA, b1, bn1_g, bn1_b, bn1_m, bn1_v, HID_C);

  // ---- layer 2: 128 -> 64 ----
  k_gemm_wmma<HID_C, OUT_C><<<mTiles, 32 * (OUT_C / 16), 0, stream>>>(bufB, W2, bufA);
  k_self<<<gNC_o, B, 0, stream>>>(bufB, bufA, deg, OUT_C);
  k_scatter<<<gS_o, B, 0, stream>>>(bufB, bufA, dinv, src, dst, OUT_C / 4, OUT_C);

  // ---- LayerNorm + pool ----
  k_ln_pool<<<(N_NODES + 7) / 8, B, 0, stream>>>(node_out, psum, pcnt, bufB, b2,
                                                 ln_g, ln_b, batch);
  k_pool_final<<<(G_GRAPHS * OUT_C + B - 1) / B, B, 0, stream>>>(graph_out, psum, pcnt);
}